// BTRLoss_28157805593332
// MI455X (gfx1250) — compile-verified
//
#include <hip/hip_runtime.h>
#include <math.h>

typedef float v2f __attribute__((ext_vector_type(2)));
typedef float v8f __attribute__((ext_vector_type(8)));

#define KSZ   16
#define TAPS  256
#define HIDN  64
#define BATCH 8
#define IMH   1024
#define IMW   1024
#define PADB  7           // (K-1)/2
#define TILE  32
#define LDSH  47          // TILE + K - 1
#define LDSW  48          // padded row stride

__device__ __forceinline__ float grid_coord(int u) {
    // jnp.linspace(-8, 8, 16)[u]
    return -8.0f + (16.0f / 15.0f) * (float)u;
}

// ---------------------------------------------------------------------------
// Kernel 1: tip MLP.  grid = 16 blocks (8 images x 2 row-halves), 256 thr.
// Layer1 scalar (K=3), Layer2 via V_WMMA_F32_16X16X4_F32, Layer3 fused via
// butterfly reduction.  Writes bh[b][256].
// ---------------------------------------------------------------------------
__global__ void __launch_bounds__(256) mlp_kernel(
    const float* __restrict__ w1, const float* __restrict__ b1,
    const float* __restrict__ w2, const float* __restrict__ b2,
    const float* __restrict__ w3, const float* __restrict__ b3,
    const int*  __restrict__ nptr, float* __restrict__ bh_out)
{
    __shared__ float w2s[HIDN * HIDN];   // 16 KB, B fragments conflict-free
    __shared__ float h1T[HIDN * 128];    // 32 KB, transposed: h1T[k][row]

    const int tid  = threadIdx.x;
    const int b    = blockIdx.x >> 1;
    const int half = blockIdx.x & 1;
    const float tval = (float)(nptr[0] * BATCH + b);

    for (int idx = tid; idx < HIDN * HIDN; idx += 256) w2s[idx] = w2[idx];

    // layer 1: 2 threads per row, 32 k's each
    {
        const int row  = tid >> 1;
        const int part = tid & 1;
        const int ig   = half * 128 + row;           // global row in [0,256)
        const float x  = grid_coord(ig >> 4);        // XF
        const float y  = grid_coord(ig & 15);        // YF
        const int k0   = part * 32;
#pragma unroll 8
        for (int k = k0; k < k0 + 32; ++k) {
            float a = fmaf(x, w1[k], fmaf(y, w1[HIDN + k],
                      fmaf(tval, w1[2 * HIDN + k], b1[k])));
            h1T[k * 128 + row] = tanhf(a);
        }
    }
    __syncthreads();

    // layer 2 + 3: wave w owns 16 local rows (one row-tile), all 4 col-tiles
    const int wave    = tid >> 5;
    const int lane    = tid & 31;
    const int m       = lane & 15;       // N (or M for A frag) within tile
    const int hhalf   = lane >> 4;       // half-wave select
    const int ksel    = hhalf << 1;      // A/B: lanes16-31 hold K+2,K+3
    const int rowbase = wave * 16;

    float rowsum[8];
#pragma unroll
    for (int r = 0; r < 8; ++r) rowsum[r] = 0.0f;

    for (int c = 0; c < 4; ++c) {
        v8f acc;
        const float bias = b2[c * 16 + m];           // col = c*16 + N
#pragma unroll
        for (int r = 0; r < 8; ++r) acc[r] = bias;   // bias rides in C

#pragma unroll
        for (int kk = 0; kk < 16; ++kk) {            // K = 64 in steps of 4
            const int k0 = kk * 4 + ksel;
            v2f a, bb;
            a.x  = h1T[k0 * 128 + rowbase + m];          // A[M=m][k0]
            a.y  = h1T[(k0 + 1) * 128 + rowbase + m];    // A[M=m][k0+1]
            bb.x = w2s[k0 * HIDN + c * 16 + m];          // B[k0][N=m]
            bb.y = w2s[(k0 + 1) * HIDN + c * 16 + m];    // B[k0+1][N=m]
            acc = __builtin_amdgcn_wmma_f32_16x16x4_f32(
                      false, a, false, bb, (short)0, acc, false, false);
        }

        const float w3v = w3[c * 16 + m];
#pragma unroll
        for (int r = 0; r < 8; ++r) {
            // D layout: VGPR r = (M = r + 8*hhalf, N = m)
            float contrib = tanhf(acc[r]) * w3v;
            contrib += __shfl_xor(contrib, 1, 32);
            contrib += __shfl_xor(contrib, 2, 32);
            contrib += __shfl_xor(contrib, 4, 32);
            contrib += __shfl_xor(contrib, 8, 32);   // sum over 16 lanes
            rowsum[r] += contrib;
        }
    }

    if (m == 0) {
        const float bias3 = b3[0];
#pragma unroll
        for (int r = 0; r < 8; ++r) {
            const int rowg = half * 128 + rowbase + r + 8 * hhalf;
            bh_out[b * TAPS + rowg] = rowsum[r] + bias3;
        }
    }
}

// ---------------------------------------------------------------------------
// Kernel 2: bh-only loss terms per image.  grid = 8, 256 thr.
// ---------------------------------------------------------------------------
__global__ void __launch_bounds__(256) sideloss_kernel(
    const float* __restrict__ bh, float* __restrict__ lossPartial)
{
    __shared__ float red[256];
    const int tid = threadIdx.x;
    const int b   = blockIdx.x;
    const float v = bh[b * TAPS + tid];
    const float x = grid_coord(tid >> 4);
    const float y = grid_coord(tid & 15);

    float s[6];
    s[0] = (v + 100.f) * (v + 100.f);     // boundary
    s[1] = v * v;                         // regularization
    s[2] = fabsf(v) * x;                  // centroid x
    s[3] = fabsf(v) * y;                  // centroid y
    s[4] = v;                             // average
    float rp = fmaxf(v, 0.f);
    s[5] = rp * rp;                       // relu(tip)^2

    float results[6];
    for (int i = 0; i < 6; ++i) {
        red[tid] = s[i];
        __syncthreads();
        for (int st = 128; st > 0; st >>= 1) {
            if (tid < st) red[tid] += red[tid + st];
            __syncthreads();
        }
        results[i] = red[0];
        __syncthreads();
    }
    red[tid] = v;
    __syncthreads();
    for (int st = 128; st > 0; st >>= 1) {
        if (tid < st) red[tid] = fmaxf(red[tid], red[tid + st]);
        __syncthreads();
    }
    if (tid == 0) {
        const float mx       = red[0];
        const float boundary = results[0] * (1.0f / 256.f);
        const float regu     = results[1];
        const float centroid = results[2] * results[2] + results[3] * results[3];
        const float avgm     = results[4] * (1.0f / 256.f);
        const float height   = results[5] * (1.0f / 256.f) + mx * mx;
        lossPartial[b] = 0.1f * boundary + height + 1e-4f * regu
                       + 0.1f * (avgm * avgm) + 1e-3f * centroid;
    }
}

// ---------------------------------------------------------------------------
// Kernel 3: erosion.  grid = (32,32,8), 256 thr, 4 px/thread.
// eroded[i,j] = -max_{u,v}( bh[u*16+v] - padded(img)[i+u, j+v] )
// ---------------------------------------------------------------------------
__global__ void __launch_bounds__(256) erosion_kernel(
    const float* __restrict__ img, const float* __restrict__ bh,
    float* __restrict__ eroded)
{
    __shared__ float tile[LDSH * LDSW];
    __shared__ float bhs[TAPS];
    const int tid = threadIdx.x;
    const int b   = blockIdx.z;
    const int ti0 = blockIdx.y * TILE;
    const int tj0 = blockIdx.x * TILE;
    const float* im = img + (size_t)b * IMH * IMW;

    bhs[tid] = bh[b * TAPS + tid];
    for (int idx = tid; idx < LDSH * LDSW; idx += 256) {
        const int r  = idx / LDSW;
        const int c  = idx - r * LDSW;
        const int gi = ti0 - PADB + r;
        const int gj = tj0 - PADB + c;
        float v = 0.f;   // zero padding (jnp.pad default)
        if (c < LDSH && gi >= 0 && gi < IMH && gj >= 0 && gj < IMW)
            v = im[gi * IMW + gj];
        tile[idx] = v;
    }
    __syncthreads();

    const int tx  = tid & 31;
    const int tyb = tid >> 5;
    float mx[4] = {-INFINITY, -INFINITY, -INFINITY, -INFINITY};
    for (int tap = 0; tap < TAPS; ++tap) {
        const float tw = bhs[tap];                 // uniform -> LDS broadcast
        const int u = tap >> 4, v = tap & 15;
        const int base = u * LDSW + tx + v;
#pragma unroll
        for (int s = 0; s < 4; ++s) {
            const float w = tile[base + (tyb + 8 * s) * LDSW];
            mx[s] = fmaxf(mx[s], tw - w);
        }
    }
    float* er = eroded + (size_t)b * IMH * IMW;
#pragma unroll
    for (int s = 0; s < 4; ++s)
        er[(size_t)(ti0 + tyb + 8 * s) * IMW + tj0 + tx] = -mx[s];
}

// ---------------------------------------------------------------------------
// Kernel 4: dilation of eroded + per-tile squared-error partial.
// rec[i,j] = max_{u,v}( bh + padded(eroded)[i+u, j+v] )
// ---------------------------------------------------------------------------
__global__ void __launch_bounds__(256) dilation_kernel(
    const float* __restrict__ eroded, const float* __restrict__ img,
    const float* __restrict__ bh, float* __restrict__ partials)
{
    __shared__ float tile[LDSH * LDSW];
    __shared__ float bhs[TAPS];
    __shared__ float red[256];
    const int tid = threadIdx.x;
    const int b   = blockIdx.z;
    const int ti0 = blockIdx.y * TILE;
    const int tj0 = blockIdx.x * TILE;
    const float* er = eroded + (size_t)b * IMH * IMW;
    const float* im = img    + (size_t)b * IMH * IMW;

    bhs[tid] = bh[b * TAPS + tid];
    for (int idx = tid; idx < LDSH * LDSW; idx += 256) {
        const int r  = idx / LDSW;
        const int c  = idx - r * LDSW;
        const int gi = ti0 - PADB + r;
        const int gj = tj0 - PADB + c;
        float v = 0.f;
        if (c < LDSH && gi >= 0 && gi < IMH && gj >= 0 && gj < IMW)
            v = er[gi * IMW + gj];
        tile[idx] = v;
    }
    __syncthreads();

    const int tx  = tid & 31;
    const int tyb = tid >> 5;
    float mx[4] = {-INFINITY, -INFINITY, -INFINITY, -INFINITY};
    for (int tap = 0; tap < TAPS; ++tap) {
        const float tw = bhs[tap];
        const int u = tap >> 4, v = tap & 15;
        const int base = u * LDSW + tx + v;
#pragma unroll
        for (int s = 0; s < 4; ++s) {
            const float w = tile[base + (tyb + 8 * s) * LDSW];
            mx[s] = fmaxf(mx[s], tw + w);
        }
    }
    float sum = 0.f;
#pragma unroll
    for (int s = 0; s < 4; ++s) {
        const float d = mx[s] - im[(size_t)(ti0 + tyb + 8 * s) * IMW + tj0 + tx];
        sum = fmaf(d, d, sum);
    }
    red[tid] = sum;
    __syncthreads();
    for (int st = 128; st > 0; st >>= 1) {
        if (tid < st) red[tid] += red[tid + st];
        __syncthreads();
    }
    if (tid == 0)
        partials[b * 1024 + blockIdx.y * 32 + blockIdx.x] = red[0];
}

// ---------------------------------------------------------------------------
// Kernel 5: deterministic final combine -> scalar loss.
// ---------------------------------------------------------------------------
__global__ void __launch_bounds__(256) final_kernel(
    const float* __restrict__ partials, const float* __restrict__ lossPartial,
    float* __restrict__ out)
{
    __shared__ float red[256];
    const int tid = threadIdx.x;
    float total = 0.f;
    for (int b = 0; b < BATCH; ++b) {
        float s = 0.f;
#pragma unroll
        for (int i = 0; i < 4; ++i) s += partials[b * 1024 + tid + i * 256];
        red[tid] = s;
        __syncthreads();
        for (int st = 128; st > 0; st >>= 1) {
            if (tid < st) red[tid] += red[tid + st];
            __syncthreads();
        }
        if (tid == 0)
            total += red[0] * (1.0f / ((float)IMH * (float)IMW)) + lossPartial[b];
        __syncthreads();
    }
    if (tid == 0) out[0] = total * (1.0f / (float)BATCH);
}

// ---------------------------------------------------------------------------
extern "C" void kernel_launch(void* const* d_in, const int* in_sizes, int n_in,
                              void* d_out, int out_size, void* d_ws, size_t ws_size,
                              hipStream_t stream)
{
    const float* images = (const float*)d_in[0];
    const float* w1     = (const float*)d_in[1];
    const float* b1     = (const float*)d_in[2];
    const float* w2     = (const float*)d_in[3];
    const float* b2     = (const float*)d_in[4];
    const float* w3     = (const float*)d_in[5];
    const float* b3     = (const float*)d_in[6];
    const int*   nptr   = (const int*)d_in[7];
    float* out = (float*)d_out;
    float* ws  = (float*)d_ws;

    float* bh          = ws;            // 8*256 floats
    float* lossPartial = ws + 2048;     // 8 floats
    float* partials    = ws + 2056;     // 8192 floats
    float* eroded      = ws + 16384;    // 8*1024*1024 floats (32 MB)

    mlp_kernel<<<16, 256, 0, stream>>>(w1, b1, w2, b2, w3, b3, nptr, bh);
    sideloss_kernel<<<8, 256, 0, stream>>>(bh, lossPartial);
    dim3 grid(IMW / TILE, IMH / TILE, BATCH);
    erosion_kernel<<<grid, 256, 0, stream>>>(images, bh, eroded);
    dilation_kernel<<<grid, 256, 0, stream>>>(eroded, images, bh, partials);
    final_kernel<<<1, 256, 0, stream>>>(partials, lossPartial, out);
}